// Pooler_46660524704167
// MI455X (gfx1250) — compile-verified
//
#include <hip/hip_runtime.h>

namespace {
constexpr int POOLED   = 7;
constexpr int NBIN     = 49;                 // 7*7
constexpr int NLVL     = 4;
constexpr int NSQ      = 4;                  // 2x2 sampling points per bin
constexpr int NSAMP    = NLVL * NSQ * NBIN;  // 784
constexpr int C        = 256;
constexpr int NPERB    = 512;                // rois per batch
constexpr int NROI     = 1024;               // total rois (B=2)
constexpr int THREADS  = 256;                // 8 wave32
constexpr int CH_SPLIT = 2;                  // channel halves per roi -> 2048 blocks
constexpr int CPB      = C / CH_SPLIT;       // 128 channels per block
}

// One block per (roi, channel-half). Phase 1 caches bilinear weights/offsets
// for all 4 levels x 49 bins x 4 subsamples in LDS (float4/int4, 25 KB).
// Phase 2: each lane owns (channel, bin) outputs; 64 gathers + 64 FMAs each.
__global__ __launch_bounds__(THREADS)
void roi_align_fpn_sum(const float* __restrict__ x0, const float* __restrict__ x1,
                       const float* __restrict__ x2, const float* __restrict__ x3,
                       const float* __restrict__ boxes, float* __restrict__ out)
{
    __shared__ float4 sw[NSAMP];   // 4 bilinear weights (mask & 1/4 mean folded in)
    __shared__ int4   so[NSAMP];   // 4 element offsets within a channel plane

    const int r   = blockIdx.x;        // roi id 0..1023
    const int b   = r / NPERB;         // batch id
    const int tid = threadIdx.x;

    // box (x1,y1,x2,y2) in image coords; broadcast load
    const float bx1 = boxes[r * 4 + 0];
    const float by1 = boxes[r * 4 + 1];
    const float bx2 = boxes[r * 4 + 2];
    const float by2 = boxes[r * 4 + 3];

    // ---- Phase 1: precompute sample weights/offsets into LDS ----
    for (int t = tid; t < NSAMP; t += THREADS) {
        const int bin = t % NBIN;
        const int u   = t / NBIN;
        const int sq  = u & 3;         // subsample 0..3
        const int lvl = u >> 2;        // level 0..3
        const int ph  = bin / POOLED;
        const int pw  = bin % POOLED;
        const int sy  = sq >> 1;
        const int sx  = sq & 1;

        const int   H     = 200 >> lvl;                 // 200,100,50,25 (W == H)
        const int   W     = H;
        const float scale = 0.25f / (float)(1 << lvl);  // .25,.125,.0625,.03125

        const float rx1 = bx1 * scale, ry1 = by1 * scale;
        const float rx2 = bx2 * scale, ry2 = by2 * scale;
        const float rw  = fmaxf(rx2 - rx1, 1.0f);
        const float rh  = fmaxf(ry2 - ry1, 1.0f);
        const float bw  = rw * (1.0f / (float)POOLED);
        const float bh  = rh * (1.0f / (float)POOLED);

        const float y = ry1 + ((float)ph + ((float)sy + 0.5f) * 0.5f) * bh;
        const float x = rx1 + ((float)pw + ((float)sx + 0.5f) * 0.5f) * bw;

        const bool ok = (y >= -1.0f) && (y <= (float)H) &&
                        (x >= -1.0f) && (x <= (float)W);

        const float yc = fminf(fmaxf(y, 0.0f), (float)(H - 1));
        const float xc = fminf(fmaxf(x, 0.0f), (float)(W - 1));
        const int   yl = (int)floorf(yc);
        const int   xl = (int)floorf(xc);
        const int   yh = min(yl + 1, H - 1);
        const int   xh = min(xl + 1, W - 1);
        const float ly = yc - (float)yl;
        const float lx = xc - (float)xl;
        const float hy = 1.0f - ly;
        const float hx = 1.0f - lx;
        const float m  = ok ? 0.25f : 0.0f;   // fold mean over 2x2 samples + mask

        sw[t] = make_float4(hy * hx * m, hy * lx * m, ly * hx * m, ly * lx * m);
        so[t] = make_int4(yl * W + xl, yl * W + xh, yh * W + xl, yh * W + xh);
    }
    __syncthreads();

    // per-level base pointers for this roi's batch
    const float* __restrict__ f0 = x0 + (size_t)b * C * 200 * 200;
    const float* __restrict__ f1 = x1 + (size_t)b * C * 100 * 100;
    const float* __restrict__ f2 = x2 + (size_t)b * C * 50 * 50;
    const float* __restrict__ f3 = x3 + (size_t)b * C * 25 * 25;

    const int c0 = blockIdx.y * CPB;   // channel-half base

    // ---- Phase 2: lanes consecutive in bin, loop channels ----
    int c   = tid / NBIN;
    int bin = tid % NBIN;
    constexpr int NOUT = CPB * NBIN;   // 6272 outputs per block

#define LEVEL_ACC(FP, HW, LVL)                                              \
    {                                                                       \
        const float* __restrict__ p = (FP) + (size_t)cc * (HW);             \
        _Pragma("unroll")                                                   \
        for (int sq = 0; sq < 4; ++sq) {                                    \
            const int    s = ((LVL)*4 + sq) * NBIN + bin;                   \
            const float4 w = sw[s];                                         \
            const int4   o = so[s];                                         \
            acc += w.x * p[o.x];                                            \
            acc += w.y * p[o.y];                                            \
            acc += w.z * p[o.z];                                            \
            acc += w.w * p[o.w];                                            \
        }                                                                   \
    }

    for (int idx = tid; idx < NOUT; idx += THREADS) {
        const int cc = c0 + c;
        float acc = 0.0f;
        LEVEL_ACC(f0, 200 * 200, 0)
        LEVEL_ACC(f1, 100 * 100, 1)
        LEVEL_ACC(f2,   50 * 50, 2)
        LEVEL_ACC(f3,   25 * 25, 3)
        out[((size_t)r * C + cc) * NBIN + bin] = acc;

        // advance (c, bin) by 256 = 5*49 + 11 without div/mod
        bin += 11;
        c   += 5;
        if (bin >= NBIN) { bin -= NBIN; c += 1; }
    }
#undef LEVEL_ACC
}

extern "C" void kernel_launch(void* const* d_in, const int* in_sizes, int n_in,
                              void* d_out, int out_size, void* d_ws, size_t ws_size,
                              hipStream_t stream) {
    const float* x0    = (const float*)d_in[0];
    const float* x1    = (const float*)d_in[1];
    const float* x2    = (const float*)d_in[2];
    const float* x3    = (const float*)d_in[3];
    const float* boxes = (const float*)d_in[4];
    float* out = (float*)d_out;

    dim3 grid(NROI, CH_SPLIT, 1);
    roi_align_fpn_sum<<<grid, THREADS, 0, stream>>>(x0, x1, x2, x3, boxes, out);
}